// LayerHypercube_A_39341900431396
// MI455X (gfx1250) — compile-verified
//
#include <hip/hip_runtime.h>

typedef float v2f __attribute__((ext_vector_type(2)));
typedef float v4f __attribute__((ext_vector_type(4)));
typedef float v8f __attribute__((ext_vector_type(8)));

#define BATCHN   2048
#define INSIZE   1024
#define OUTSIZE  1024
#define FM       16
#define NBITS    10
#define O_TILE   32
#define STR      36   // padded LDS row stride (floats): conflict-free + 16B aligned

// Each workgroup: 16 batch rows x 32 output columns x all 16 feature maps.
// 8 waves; wave w handles o_local = 4w .. 4w+3. Per o: one 16x16 (batch x
// feature) GEMM with K=12 (j padded from 10) via 3x V_WMMA_F32_16X16X4_F32.
// x[b,o] / bias[f,o] epilogue tiles are prefetched to LDS with
// global_load_async_to_lds_b128 (ASYNCcnt), overlapped with the WMMA phase.
// Writeback is fully vectorized: ds_load_b128 + global_store_b128, lanes
// contiguous in o => perfectly coalesced output stores (the HBM bottleneck).
__global__ __launch_bounds__(256)
void hypercube_wmma_kernel(const float* __restrict__ x,
                           const float* __restrict__ w,
                           const float* __restrict__ bias,
                           float* __restrict__ out)
{
    __shared__ float stage[16 * 16 * STR];  // [b_local*16 + f][STR]
    __shared__ float xstage[16 * O_TILE];   // x[b0+bl][o_base+oo]
    __shared__ float bstage[FM * O_TILE];   // bias[f][o_base+oo]

    const int tid    = threadIdx.x;
    const int lane   = tid & 31;
    const int wv     = tid >> 5;        // wave id 0..7
    const int half   = lane >> 4;       // 0: lanes 0-15, 1: lanes 16-31
    const int m      = lane & 15;       // A: row M (batch); B/D: col N (feature)

    const int o_base = blockIdx.x * O_TILE;
    const int b0     = blockIdx.y * 16;

    // ---- async prefetch of epilogue tiles into LDS: one b128 per thread.
    // threads 0..127 -> xstage (16x32 f32), threads 128..255 -> bstage.
    {
        const bool isx = tid < 128;
        const int  idx4 = (tid & 127) * 4;          // float index 0..508
        const int  bl   = idx4 >> 5;                // row 0..15 (batch or feature)
        const int  oo   = idx4 & 31;                // 16B-aligned column quad
        float* larr = isx ? xstage : bstage;
        const float* g = isx ? (x    + (size_t)(b0 + bl) * INSIZE + o_base + oo)
                             : (bias + (size_t)bl * OUTSIZE       + o_base + oo);
        unsigned loff = (unsigned)(uintptr_t)(&larr[idx4]); // low 32b = LDS offset
        asm volatile("global_load_async_to_lds_b128 %0, %1, off"
                     :: "v"(loff), "v"(g) : "memory");
    }

    const float* xrow = x + (size_t)(b0 + m) * INSIZE;     // A operand row (M = m)
    const float* wrow = w + (size_t)m * (NBITS * OUTSIZE); // B operand col (f = m)

    // per-lane K assignment inside each 16x16x4 step:
    // lanes 0-15 hold K = {0,1}, lanes 16-31 hold K = {2,3} -> j = 4k + 2*half + {0,1}
    const int j0 = 2 * half;

    for (int ol = 0; ol < 4; ++ol) {
        const int o_local = wv * 4 + ol;
        const int o = o_base + o_local;

        v2f a0, a1, a2, bb0, bb1, bb2;
        // k = 0: j = j0, j0+1   (0/1 or 2/3)
        a0.x  = xrow[o ^ (1 << (j0 + 0))];
        a0.y  = xrow[o ^ (1 << (j0 + 1))];
        bb0.x = wrow[(j0 + 0) * OUTSIZE + o];
        bb0.y = wrow[(j0 + 1) * OUTSIZE + o];
        // k = 1: j = j0+4, j0+5 (4/5 or 6/7)
        a1.x  = xrow[o ^ (1 << (j0 + 4))];
        a1.y  = xrow[o ^ (1 << (j0 + 5))];
        bb1.x = wrow[(j0 + 4) * OUTSIZE + o];
        bb1.y = wrow[(j0 + 5) * OUTSIZE + o];
        // k = 2: rows K=8,9 (j=8,9) are real; rows K=10,11 (lanes 16-31) are
        // padding. Load always-valid j=8,9 addresses and zero only the B
        // operand with a post-load select (no EXEC divergence, product == 0).
        a2.x = xrow[o ^ (1 << 8)];
        a2.y = xrow[o ^ (1 << 9)];
        {
            const float b2x = wrow[8 * OUTSIZE + o];
            const float b2y = wrow[9 * OUTSIZE + o];
            bb2.x = half ? 0.0f : b2x;
            bb2.y = half ? 0.0f : b2y;
        }

        v8f c = {0.f, 0.f, 0.f, 0.f, 0.f, 0.f, 0.f, 0.f};
        c = __builtin_amdgcn_wmma_f32_16x16x4_f32(false, a0, false, bb0,
                                                  (short)0, c, false, false);
        c = __builtin_amdgcn_wmma_f32_16x16x4_f32(false, a1, false, bb1,
                                                  (short)0, c, false, false);
        c = __builtin_amdgcn_wmma_f32_16x16x4_f32(false, a2, false, bb2,
                                                  (short)0, c, false, false);

        // D layout: VGPR r, lane l -> M = r + 8*half (batch), N = l&15 (feature)
        #pragma unroll
        for (int r = 0; r < 8; ++r) {
            const int bl = r + 8 * half;
            stage[(bl * 16 + m) * STR + o_local] = c[r];
        }
    }

    // ensure this wave's async LDS writes have landed, then workgroup barrier
#if __has_builtin(__builtin_amdgcn_s_wait_asynccnt)
    __builtin_amdgcn_s_wait_asynccnt(0);
#else
    asm volatile("s_wait_asynccnt 0" ::: "memory");
#endif
    __syncthreads();

    // Vectorized coalesced writeback + epilogue:
    //   out[b, f*1024+o..o+3] = D + bias[f,o..] + x[b,o..]   (float4 everywhere)
    // thread t: o-quad = (t&7)*4; 8 (f,b_local) pairs from (t>>3)*8 + i.
    const int oq = (tid & 7) * 4;       // o_local of this thread's quad
    const int g  = tid >> 3;            // 0..31
    #pragma unroll
    for (int i = 0; i < 8; ++i) {
        const int q  = g * 8 + i;       // 0..255 -> all (f, b_local) pairs
        const int f  = q >> 4;
        const int bl = q & 15;
        const v4f s  = *(const v4f*)&stage[(bl * 16 + f) * STR + oq];
        const v4f bv = *(const v4f*)&bstage[f * O_TILE + oq];
        const v4f xv = *(const v4f*)&xstage[bl * O_TILE + oq];
        v4f v = s + bv + xv;
        *(v4f*)&out[(size_t)(b0 + bl) * (FM * OUTSIZE) + f * OUTSIZE + o_base + oq] = v;
    }
}

extern "C" void kernel_launch(void* const* d_in, const int* in_sizes, int n_in,
                              void* d_out, int out_size, void* d_ws, size_t ws_size,
                              hipStream_t stream) {
    (void)in_sizes; (void)n_in; (void)d_ws; (void)ws_size; (void)out_size;
    const float* x    = (const float*)d_in[0];   // (2048, 1024) f32
    const float* wts  = (const float*)d_in[1];   // (16, 10, 1024) f32
    const float* bias = (const float*)d_in[2];   // (16, 1024) f32
    // d_in[3] = fm (int32) is unused: fm[f,j,o] == o ^ (1<<j), computed inline.
    float* out = (float*)d_out;                  // (2048, 16384) f32

    dim3 grid(OUTSIZE / O_TILE, BATCHN / 16);    // (32, 128)
    dim3 block(256);
    hypercube_wmma_kernel<<<grid, block, 0, stream>>>(x, wts, bias, out);
}